// VoxelHashTable_23338852286543
// MI455X (gfx1250) — compile-verified
//
#include <hip/hip_runtime.h>
#include <stdint.h>

// VoxelHashTable (instant-NGP hash-grid encode), MI455X / gfx1250.
// Memory-bound gather workload: WMMA is inapplicable (per-point feature
// matrices), so we use the CDNA5 async path: global_load_async_to_lds_b128
// gathers, double-buffered per wave, drained with s_wait_asynccnt.
// Feature gathers stay RT-cached (working set ~60MB << 192MB L2); the
// touch-once output/point streams are non-temporal so they don't evict it.

typedef __attribute__((ext_vector_type(4))) float v4f;  // native vector for NT builtins

namespace {

constexpr int      kPts    = 262144;
constexpr int      kUnits  = kPts * 2;          // (point, level) units
constexpr int      kBlock  = 256;               // 8 waves of 32
constexpr int      kIters  = 8;                 // units streamed per thread
constexpr int      kGrid   = kUnits / kBlock;   // 2048 blocks (524288 threads)
constexpr int      kUStep  = kUnits / kIters;   // 65536 unit stride per iter
constexpr uint32_t kMask   = (1u << 21) - 1;    // BUCKETS = 2^21

// Compute the 8 corner gather addresses + trilinear weights for one
// (point, level) unit. All hash math in uint32: mod 2^21 == AND(0x1FFFFF),
// and the low 21 bits of the int64 products match int32 wraparound exactly.
__device__ __forceinline__ void addr_and_weights(
    const float* __restrict__ pts, const float* __restrict__ feats,
    int unit, int chunk, float res, uint64_t gaddr[8], float w[8])
{
  const int   p  = unit >> 1;
  const float px = __builtin_nontemporal_load(&pts[p * 3 + 0]);
  const float py = __builtin_nontemporal_load(&pts[p * 3 + 1]);
  const float pz = __builtin_nontemporal_load(&pts[p * 3 + 2]);
  // Exact (correctly rounded) f32 division to match the reference's floor()
  // decisions bit-for-bit at cell boundaries.
  const float qx = px / res, qy = py / res, qz = pz / res;
  const float bx = floorf(qx), by = floorf(qy), bz = floorf(qz);
  const float fx = qx - bx, fy = qy - by, fz = qz - bz;
  const int ix = (int)bx, iy = (int)by, iz = (int)bz;
  const uint32_t hx0 = (uint32_t)ix * 73856093u;
  const uint32_t hx1 = hx0 + 73856093u;
  const uint32_t hy0 = (uint32_t)iy * 19349669u;
  const uint32_t hy1 = hy0 + 19349669u;
  const uint32_t hz0 = (uint32_t)iz * 83492791u;
  const uint32_t hz1 = hz0 + 83492791u;
  const uint64_t base = (uint64_t)(uintptr_t)feats + (uint32_t)(chunk * 16);
#pragma unroll
  for (int c = 0; c < 8; ++c) {
    const uint32_t h =
        ((c & 1) ? hx1 : hx0) + ((c & 2) ? hy1 : hy0) + ((c & 4) ? hz1 : hz0);
    const uint64_t vid = (uint64_t)(h & kMask);
    gaddr[c] = base + vid * 128u;  // row stride: 32 floats = 128 B
    const float wx = (c & 1) ? fx : 1.0f - fx;
    const float wy = (c & 2) ? fy : 1.0f - fy;
    const float wz = (c & 4) ? fz : 1.0f - fz;
    w[c] = wx * wy * wz;
  }
}

}  // namespace

__global__ __launch_bounds__(kBlock) void hashgrid_gather_kernel(
    const float* __restrict__ pts,
    const float* __restrict__ feats0,
    const float* __restrict__ feats1,
    float* __restrict__ out)
{
  // Per-wave private staging: [wave][buf][corner][lane] (8 KB per wave).
  __shared__ v4f stage[kBlock / 32][2][8][32];

  const int tid   = blockIdx.x * kBlock + (int)threadIdx.x;
  const int lane  = threadIdx.x & 31;
  const int wave  = threadIdx.x >> 5;
  const int chunk = tid & 7;       // which float4 of the 32-dim feature row
  const int ubase = tid >> 3;      // unit handled at iteration 0
  const int level = ubase & 1;     // constant across iterations (uStep even)
  const float res = level ? 0.12f : 0.24f;
  const float* __restrict__ feats = level ? feats1 : feats0;

  const uint32_t lds0 = (uint32_t)(uintptr_t)&stage[wave][0][0][lane];
  const uint32_t lds1 = (uint32_t)(uintptr_t)&stage[wave][1][0][lane];

  // Prologue: issue async gathers for iteration 0 into buffer 0.
  uint64_t ga[8];
  float    w[8];
  addr_and_weights(pts, feats, ubase, chunk, res, ga, w);
#pragma unroll
  for (int c = 0; c < 8; ++c) {
    const uint32_t dst = lds0 + (uint32_t)(c * 512);  // corner stride 32*16 B
    asm volatile("global_load_async_to_lds_b128 %0, %1, off"
                 :: "v"(dst), "v"(ga[c]) : "memory");
  }

  v4f* __restrict__ out4 = (v4f*)out;

#pragma unroll
  for (int j = 0; j < kIters; ++j) {
    const int u = j * kUStep + ubase;
    float w_next[8];
    if (j + 1 < kIters) {
      // Issue the next batch into the other buffer, then drain this batch.
      uint64_t ga_next[8];
      addr_and_weights(pts, feats, u + kUStep, chunk, res, ga_next, w_next);
      const uint32_t ldsn = ((j + 1) & 1) ? lds1 : lds0;
#pragma unroll
      for (int c = 0; c < 8; ++c) {
        const uint32_t dst = ldsn + (uint32_t)(c * 512);
        asm volatile("global_load_async_to_lds_b128 %0, %1, off"
                     :: "v"(dst), "v"(ga_next[c]) : "memory");
      }
      // Async loads complete in order: <=8 outstanding => batch j is done.
      asm volatile("s_wait_asynccnt 0x8" ::: "memory");
    } else {
      asm volatile("s_wait_asynccnt 0x0" ::: "memory");
    }

    const int buf = j & 1;
    v4f acc = {0.f, 0.f, 0.f, 0.f};
#pragma unroll
    for (int c = 0; c < 8; ++c) {
      const v4f v = stage[wave][buf][c][lane];
      acc.x = fmaf(w[c], v.x, acc.x);
      acc.y = fmaf(w[c], v.y, acc.y);
      acc.z = fmaf(w[c], v.z, acc.z);
      acc.w = fmaf(w[c], v.w, acc.w);
    }
    // unit u owns out floats [u*32, u*32+32) == out4[u*8 .. u*8+8).
    // Streaming, touch-once output: non-temporal so it doesn't evict the
    // L2-resident feature working set.
    __builtin_nontemporal_store(acc, &out4[u * 8 + chunk]);

    if (j + 1 < kIters) {
#pragma unroll
      for (int c = 0; c < 8; ++c) w[c] = w_next[c];
    }
  }
}

extern "C" void kernel_launch(void* const* d_in, const int* in_sizes, int n_in,
                              void* d_out, int out_size, void* d_ws, size_t ws_size,
                              hipStream_t stream) {
  (void)in_sizes; (void)n_in; (void)out_size; (void)d_ws; (void)ws_size;
  const float* pts = (const float*)d_in[0];
  const float* f0  = (const float*)d_in[1];
  const float* f1  = (const float*)d_in[2];
  float*       out = (float*)d_out;
  hipLaunchKernelGGL(hashgrid_gather_kernel, dim3(kGrid), dim3(kBlock), 0, stream,
                     pts, f0, f1, out);
}